// SortPool_38122129719446
// MI455X (gfx1250) — compile-verified
//
#include <hip/hip_runtime.h>
#include <hip/hip_bf16.h>

// ---------------- problem constants (match setup_inputs) ----------------
#define BG   1024          // graphs
#define NN   128           // nodes per graph
#define VN   (BG * NN)     // total nodes = 131072
#define HD   128           // feature / hidden dim
#define EPG  2048          // edges per graph (E / B), edges contiguous per graph
#define KTOP 30
#define CONV_O 32
#define CONV_W 3
#define CONV_T (KTOP - CONV_W + 1)   // 28
#define FLAT   (CONV_O * CONV_T)     // 896
#define NCLS   16

typedef float v2f __attribute__((ext_vector_type(2)));
typedef float v8f __attribute__((ext_vector_type(8)));

// =======================================================================
// Kernel 1: per-graph mean aggregation.  One workgroup per graph.
// feat + acc live in LDS (128 KB; CDNA5 WGP has 320 KB LDS).
// =======================================================================
__global__ __launch_bounds__(256) void sage_aggregate(
    const float* __restrict__ X, const int* __restrict__ src,
    const int* __restrict__ dst, float* __restrict__ meanOut)
{
  __shared__ float feat[NN * HD];   // 64 KB
  __shared__ float acc [NN * HD];   // 64 KB
  __shared__ float deg [NN];

  const int g = blockIdx.x;
  const int t = threadIdx.x;        // 256 threads
  const float* xg = X + (size_t)g * NN * HD;

  // load node features, zero accumulator (float4 vectorized)
  for (int i = t; i < NN * HD / 4; i += 256) {
    ((float4*)feat)[i] = ((const float4*)xg)[i];
    ((float4*)acc)[i]  = make_float4(0.f, 0.f, 0.f, 0.f);
  }
  if (t < NN) deg[t] = 0.f;
  __syncthreads();

  // 64 edge-lanes x 4 feature-groups of 32
  const int ebase = g * EPG;
  const int eg = t >> 2;            // 0..63
  const int fg = t & 3;             // 0..3
  for (int e = eg; e < EPG; e += 64) {
    const int s = src[ebase + e] & (NN - 1);   // local node id (g*128 + r)
    const int d = dst[ebase + e] & (NN - 1);
    const float* fs = feat + s * HD + fg * 32;
    float*       ad = acc  + d * HD + fg * 32;
    if (fg == 0) atomicAdd(&deg[d], 1.0f);
#pragma unroll
    for (int f = 0; f < 32; ++f) atomicAdd(&ad[f], fs[f]);
  }
  __syncthreads();

  // mean = acc / max(deg,1)
  float* mg = meanOut + (size_t)g * NN * HD;
  for (int i = t; i < NN * HD; i += 256) {
    const float c = deg[i >> 7];
    mg[i] = acc[i] / (c > 1.f ? c : 1.f);
  }
}

// =======================================================================
// Kernel 2: H = relu(Aagg @ Wl + Aself @ Wr + bias) via f32 WMMA 16x16x4.
// Block = 8 waves = 128 rows x 128 cols. Wave w owns rows [w*16, w*16+16)
// and all 8 column tiles (8 accumulators). Weights [Wl;Wr] staged in LDS
// once per block, packed as (W[k][col], W[k+1][col]) float2 pairs so each
// B fragment is one ds_load_b64.
// Fragment layouts per CDNA5 ISA 7.12.2:
//   A 16x4 : lane<16 -> (M=lane, K={0,1}); lane>=16 -> (M=lane-16, K={2,3})
//   B 4x16 : lane<16 -> (N=lane, K={0,1}); lane>=16 -> (N=lane-16, K={2,3})
//   C/D    : VGPR v -> M = v + 8*(lane>=16), N = lane&15
// =======================================================================
__global__ __launch_bounds__(256) void sage_gemm(
    const float* __restrict__ Aagg, const float* __restrict__ Aself,
    const float* __restrict__ Wl,   const float* __restrict__ Wr,
    const float* __restrict__ bias, float* __restrict__ out)
{
  // wp[kk*128+col] = (Wc[2kk][col], Wc[2kk+1][col]); Wc = [Wl;Wr] (256x128)
  __shared__ float2 wp[128 * HD];   // 128 KB

  const int t = threadIdx.x;
  for (int idx = t; idx < 128 * HD; idx += 256) {
    const int kk  = idx >> 7;
    const int col = idx & (HD - 1);
    const float* W = (kk < 64) ? Wl : Wr;
    const int k = (kk & 63) * 2;
    float2 v;
    v.x = W[(size_t)k       * HD + col];   // coalesced across col
    v.y = W[(size_t)(k + 1) * HD + col];
    wp[idx] = v;
  }
  __syncthreads();

  const int wave = t >> 5;                 // 0..7
  const int lane = t & 31;
  const int m    = lane & 15;
  const int hi   = lane >> 4;              // 0 or 1
  const int kb   = hi * 2;                 // K sub-offset {0,2}

  const int rowBase = blockIdx.x * 128 + wave * 16;
  const float* arow1 = Aagg  + (size_t)(rowBase + m) * HD;
  const float* arow2 = Aself + (size_t)(rowBase + m) * HD;

  v8f acc0 = {}, acc1 = {}, acc2 = {}, acc3 = {};
  v8f acc4 = {}, acc5 = {}, acc6 = {}, acc7 = {};

#define GEMM_STEP(AROW, KOFF)                                              \
  {                                                                        \
    v2f a = *(const v2f*)((AROW) + k0 + kb);                               \
    const float2* bp = wp + ((k0 >> 1) + hi + (KOFF)) * HD + m;            \
    float2 bb; v2f b;                                                      \
    bb = bp[0];   b.x = bb.x; b.y = bb.y;                                  \
    acc0 = __builtin_amdgcn_wmma_f32_16x16x4_f32(false, a, false, b, (short)0, acc0, false, false); \
    bb = bp[16];  b.x = bb.x; b.y = bb.y;                                  \
    acc1 = __builtin_amdgcn_wmma_f32_16x16x4_f32(false, a, false, b, (short)0, acc1, false, false); \
    bb = bp[32];  b.x = bb.x; b.y = bb.y;                                  \
    acc2 = __builtin_amdgcn_wmma_f32_16x16x4_f32(false, a, false, b, (short)0, acc2, false, false); \
    bb = bp[48];  b.x = bb.x; b.y = bb.y;                                  \
    acc3 = __builtin_amdgcn_wmma_f32_16x16x4_f32(false, a, false, b, (short)0, acc3, false, false); \
    bb = bp[64];  b.x = bb.x; b.y = bb.y;                                  \
    acc4 = __builtin_amdgcn_wmma_f32_16x16x4_f32(false, a, false, b, (short)0, acc4, false, false); \
    bb = bp[80];  b.x = bb.x; b.y = bb.y;                                  \
    acc5 = __builtin_amdgcn_wmma_f32_16x16x4_f32(false, a, false, b, (short)0, acc5, false, false); \
    bb = bp[96];  b.x = bb.x; b.y = bb.y;                                  \
    acc6 = __builtin_amdgcn_wmma_f32_16x16x4_f32(false, a, false, b, (short)0, acc6, false, false); \
    bb = bp[112]; b.x = bb.x; b.y = bb.y;                                  \
    acc7 = __builtin_amdgcn_wmma_f32_16x16x4_f32(false, a, false, b, (short)0, acc7, false, false); \
  }

#pragma unroll 2
  for (int k0 = 0; k0 < HD; k0 += 4) GEMM_STEP(arow1, 0)
#pragma unroll 2
  for (int k0 = 0; k0 < HD; k0 += 4) GEMM_STEP(arow2, 64)
#undef GEMM_STEP

  // epilogue: bias + relu + store
  const int rbase = rowBase + hi * 8;
  v8f accs[8] = {acc0, acc1, acc2, acc3, acc4, acc5, acc6, acc7};
#pragma unroll
  for (int ct = 0; ct < 8; ++ct) {
    const int col = ct * 16 + m;
    const float bv = bias[col];
#pragma unroll
    for (int v = 0; v < 8; ++v) {
      float val = accs[ct][v] + bv;
      val = val > 0.f ? val : 0.f;        // ReLU (all 3 layers)
      out[(size_t)(rbase + v) * HD + col] = val;
    }
  }
}

// =======================================================================
// Kernel 3: SortPool (top-30 by channel 127, desc) + Conv1d + MLP + logsoftmax.
// One workgroup (256 threads) per graph.
// =======================================================================
__global__ __launch_bounds__(256) void sortpool_head(
    const float* __restrict__ H,
    const float* __restrict__ conv_w, const float* __restrict__ conv_b,
    const float* __restrict__ W1, const float* __restrict__ b1,
    const float* __restrict__ W2, const float* __restrict__ b2,
    float* __restrict__ out)
{
  __shared__ float vals[NN];
  __shared__ int   idxs[NN];
  __shared__ float xs[KTOP * HD];   // 15 KB
  __shared__ float y[FLAT];         // 3.5 KB
  __shared__ float z[HD];
  __shared__ float logits[NCLS];
  __shared__ float lse;

  const int g = blockIdx.x;
  const int t = threadIdx.x;
  const float* hg = H + (size_t)g * NN * HD;

  if (t < NN) { vals[t] = hg[t * HD + (HD - 1)]; idxs[t] = t; }
  __syncthreads();

  // bitonic sort, total order: value desc, index asc (== stable argsort of -x)
  for (int k = 2; k <= NN; k <<= 1) {
    for (int j = k >> 1; j > 0; j >>= 1) {
      if (t < NN) {
        const int ixj = t ^ j;
        if (ixj > t) {
          const bool dirDesc = ((t & k) == 0);
          const float va = vals[t], vb = vals[ixj];
          const int   ia = idxs[t], ib = idxs[ixj];
          const bool aBeforeB = (va > vb) || (va == vb && ia < ib);
          if (dirDesc ? !aBeforeB : aBeforeB) {
            vals[t] = vb; vals[ixj] = va;
            idxs[t] = ib; idxs[ixj] = ia;
          }
        }
      }
      __syncthreads();
    }
  }

  // gather top-30 rows
  for (int i = t; i < KTOP * HD; i += 256) {
    const int tt = i >> 7, c = i & (HD - 1);
    xs[i] = hg[(size_t)idxs[tt] * HD + c];
  }
  __syncthreads();

  // Conv1d: y[o*28+tt] = relu(conv_b[o] + sum_{w,c} xs[tt+w][c]*conv_w[o][c][w])
  for (int i = t; i < FLAT; i += 256) {
    const int o = i / CONV_T, tt = i % CONV_T;
    float s = conv_b[o];
#pragma unroll
    for (int w = 0; w < CONV_W; ++w) {
      const float* xr = xs + (tt + w) * HD;
      const float* wr = conv_w + (size_t)o * HD * CONV_W + w;
      for (int c = 0; c < HD; ++c) s += xr[c] * wr[c * CONV_W];
    }
    y[i] = s > 0.f ? s : 0.f;
  }
  __syncthreads();

  // dense1 (896 -> 128) + relu
  if (t < HD) {
    float s = b1[t];
    for (int i = 0; i < FLAT; ++i) s += y[i] * W1[(size_t)i * HD + t];
    z[t] = s > 0.f ? s : 0.f;
  }
  __syncthreads();

  // dense2 (128 -> 16)
  if (t < NCLS) {
    float s = b2[t];
    for (int j = 0; j < HD; ++j) s += z[j] * W2[j * NCLS + t];
    logits[t] = s;
  }
  __syncthreads();

  if (t == 0) {
    float mx = logits[0];
    for (int c = 1; c < NCLS; ++c) mx = fmaxf(mx, logits[c]);
    float se = 0.f;
    for (int c = 0; c < NCLS; ++c) se += expf(logits[c] - mx);
    lse = mx + logf(se);
  }
  __syncthreads();
  if (t < NCLS) out[(size_t)g * NCLS + t] = logits[t] - lse;
}

// =======================================================================
// Host-side launch. Inputs in setup_inputs() dict order:
// 0:x 1:src 2:dst 3:batch 4:k 5:Wl1 6:bl1 7:Wr1 8:Wl2 9:bl2 10:Wr2
// 11:Wl3 12:bl3 13:Wr3 14:conv_w 15:conv_b 16:W1 17:b1 18:W2 19:b2
// =======================================================================
extern "C" void kernel_launch(void* const* d_in, const int* in_sizes, int n_in,
                              void* d_out, int out_size, void* d_ws, size_t ws_size,
                              hipStream_t stream) {
  (void)in_sizes; (void)n_in; (void)out_size; (void)ws_size;
  const float* x      = (const float*)d_in[0];
  const int*   src    = (const int*)d_in[1];
  const int*   dst    = (const int*)d_in[2];
  const float* Wl1    = (const float*)d_in[5];
  const float* bl1    = (const float*)d_in[6];
  const float* Wr1    = (const float*)d_in[7];
  const float* Wl2    = (const float*)d_in[8];
  const float* bl2    = (const float*)d_in[9];
  const float* Wr2    = (const float*)d_in[10];
  const float* Wl3    = (const float*)d_in[11];
  const float* bl3    = (const float*)d_in[12];
  const float* Wr3    = (const float*)d_in[13];
  const float* conv_w = (const float*)d_in[14];
  const float* conv_b = (const float*)d_in[15];
  const float* W1     = (const float*)d_in[16];
  const float* b1     = (const float*)d_in[17];
  const float* W2     = (const float*)d_in[18];
  const float* b2     = (const float*)d_in[19];

  float* ws   = (float*)d_ws;
  float* mean = ws;                                // VN*HD
  float* h1   = ws + (size_t)VN * HD;              // VN*HD
  float* h2   = h1 + (size_t)VN * HD;              // VN*HD

  const int gemmBlocks = VN / 128;                 // 1024 blocks of 8 waves

  // layer 1: x -> h1
  sage_aggregate<<<BG, 256, 0, stream>>>(x, src, dst, mean);
  sage_gemm<<<gemmBlocks, 256, 0, stream>>>(mean, x, Wl1, Wr1, bl1, h1);
  // layer 2: h1 -> h2
  sage_aggregate<<<BG, 256, 0, stream>>>(h1, src, dst, mean);
  sage_gemm<<<gemmBlocks, 256, 0, stream>>>(mean, h1, Wl2, Wr2, bl2, h2);
  // layer 3: h2 -> h1
  sage_aggregate<<<BG, 256, 0, stream>>>(h2, src, dst, mean);
  sage_gemm<<<gemmBlocks, 256, 0, stream>>>(mean, h2, Wl3, Wr3, bl3, h1);
  // head
  sortpool_head<<<BG, 256, 0, stream>>>(h1, conv_w, conv_b, W1, b1, W2, b2,
                                        (float*)d_out);
}